// STMP_CRL_58866821759225
// MI455X (gfx1250) — compile-verified
//
#include <hip/hip_runtime.h>
#include <hip/hip_bf16.h>

// ---------------------------------------------------------------------------
// Problem constants (from reference)
// ---------------------------------------------------------------------------
constexpr size_t kNU  = 100000;
constexpr size_t kNI  = 50000;
constexpr size_t kN   = kNU + kNI;     // 150000 graph nodes
constexpr int    kD   = 64;
constexpr int    kB   = 2048;
constexpr int    kT   = 200;
constexpr size_t kNNZ = 2000000;
constexpr float  kEPS = 1e-8f;

typedef __attribute__((ext_vector_type(2))) float v2f;
typedef __attribute__((ext_vector_type(8))) float v8f;

// 16-byte vector type matching the async-to-LDS builtin's pointee type
typedef int v4i_vs __attribute__((vector_size(16)));
typedef v4i_vs __attribute__((address_space(1)))* g128p;   // global src
typedef v4i_vs __attribute__((address_space(3)))* l128p;   // LDS dst

// ---------------------------------------------------------------------------
// fp32 WMMA: D(16x16) = A(16x4) * B(4x16) + C   -> v_wmma_f32_16x16x4_f32
// ---------------------------------------------------------------------------
__device__ __forceinline__ v8f wmma_f32x4(v2f a, v2f b, v8f c) {
  return __builtin_amdgcn_wmma_f32_16x16x4_f32(false, a, false, b,
                                               (short)0, c, false, false);
}

// ---------------------------------------------------------------------------
// Generic helpers
// ---------------------------------------------------------------------------
__global__ void zero_kernel(float* __restrict__ p, size_t n) {
  for (size_t i = (size_t)blockIdx.x * blockDim.x + threadIdx.x; i < n;
       i += (size_t)gridDim.x * blockDim.x)
    p[i] = 0.0f;
}

// emb0 = concat(user_table, item_table)
__global__ void concat_kernel(const float* __restrict__ ut,
                              const float* __restrict__ it,
                              float* __restrict__ cur) {
  const size_t total = kN * (size_t)kD;
  const size_t split = kNU * (size_t)kD;
  for (size_t i = (size_t)blockIdx.x * blockDim.x + threadIdx.x; i < total;
       i += (size_t)gridDim.x * blockDim.x)
    cur[i] = (i < split) ? ut[i] : it[i - split];
}

// next[rows[e]] += vals[e] * cur[cols[e]]  (one 32-thread group per nnz,
// each thread owns a float2 of the 64-wide row -> coalesced 256B per nnz)
__global__ void scatter_kernel(const int* __restrict__ rows,
                               const int* __restrict__ cols,
                               const float* __restrict__ vals,
                               const float* __restrict__ cur,
                               float* __restrict__ nxt) {
  size_t g = (size_t)blockIdx.x * blockDim.x + threadIdx.x;
  size_t e = g >> 5;
  if (e >= kNNZ) return;
  int d2 = (int)(g & 31) * 2;
  int rr = rows[e];
  int cc = cols[e];
  float v = vals[e];
  float2 src = *reinterpret_cast<const float2*>(cur + (size_t)cc * kD + d2);
  atomicAdd(nxt + (size_t)rr * kD + d2,     v * src.x);
  atomicAdd(nxt + (size_t)rr * kD + d2 + 1, v * src.y);
}

// layer-0 contribution: gcn_u[b,:] = user_table[ui[b],:], gcn_i likewise.
// The running mean only ever gets READ at the gathered rows, so we never
// materialize a full-N mean buffer (saves ~230MB of HBM traffic + 38MB ws).
__global__ void gather_init_kernel(const int* __restrict__ ui,
                                   const int* __restrict__ ii,
                                   const float* __restrict__ ut,
                                   const float* __restrict__ it,
                                   float* __restrict__ gu,
                                   float* __restrict__ gi) {
  int g = blockIdx.x * blockDim.x + threadIdx.x;
  if (g >= kB * kD) return;
  int b = g >> 6, d = g & 63;
  gu[g] = ut[(size_t)ui[b] * kD + d];
  gi[g] = it[(size_t)ii[b] * kD + d];
}

// gcn_u[b,:] = (gcn_u[b,:] + layer[ui[b],:]) * scale   (scale=1, last=0.25)
__global__ void gather_acc_kernel(const int* __restrict__ ui,
                                  const int* __restrict__ ii,
                                  const float* __restrict__ layer,
                                  float* __restrict__ gu,
                                  float* __restrict__ gi,
                                  float scale) {
  int g = blockIdx.x * blockDim.x + threadIdx.x;
  if (g >= kB * kD) return;
  int b = g >> 6, d = g & 63;
  gu[g] = (gu[g] + layer[(size_t)ui[b] * kD + d]) * scale;
  gi[g] = (gi[g] + layer[(kNU + (size_t)ii[b]) * kD + d]) * scale;
}

// ---------------------------------------------------------------------------
// GRU. Each wave owns a 16-row batch tile.
//  * W_ih / W_hh are copied ONCE per block into LDS (stride-68 padded rows:
//    bank = (4*row + k) % 64 -> conflict-free for the 16x4 WMMA A/B fetch
//    pattern). Preload uses GLOBAL_LOAD_ASYNC_TO_LDS_B128 (ASYNCcnt path)
//    when the toolchain exposes it, else a plain cooperative copy.
//  * h lives in LDS in the same padded layout; same-wave DS ordering makes
//    the read(A-layout) -> write(D-layout) recurrence safe without barriers.
//  * r/z gates use sigmoid(gi+gh), so those 8 column-tiles accumulate X*Wih^T
//    and H*Whh^T into a SHARED accumulator; only the n-gate keeps them split.
//  * all 16 x-chunks of step t are issued as one load batch and event_ids of
//    step t+1 is prefetched during step t.
// ---------------------------------------------------------------------------
__global__ __launch_bounds__(128) void gru_kernel(
    const int* __restrict__ event_ids,    // [T,B]
    const int* __restrict__ event_lens,   // [B]
    const float* __restrict__ event_table,// [NI,64]
    const float* __restrict__ W_ih,       // [192,64]
    const float* __restrict__ W_hh,       // [192,64]
    const float* __restrict__ b_ih,       // [192]
    const float* __restrict__ b_hh,       // [192]
    float* __restrict__ gru_out)          // [B,64]
{
  constexpr int HS = 68;                  // padded LDS row stride (floats)
  extern __shared__ float smem[];
  float* sWi  = smem;                     // [192*68]
  float* sWh  = smem + 192 * HS;          // [192*68]
  float* sHal = smem + 2 * 192 * HS;      // [4][16*68]

  const int tid = threadIdx.x;

  // ---- weight preload into padded LDS rows ----
#if __has_builtin(__builtin_amdgcn_global_load_async_to_lds_b128)
  for (int q = tid; q < 192 * 16; q += 128) {   // one 16B chunk per lane/iter
    int w = q >> 4, k4 = (q & 15) << 2;         // row, k-offset (floats)
    __builtin_amdgcn_global_load_async_to_lds_b128(
        (g128p)(W_ih + (size_t)w * kD + k4),
        (l128p)(sWi + w * HS + k4), 0, 0);
    __builtin_amdgcn_global_load_async_to_lds_b128(
        (g128p)(W_hh + (size_t)w * kD + k4),
        (l128p)(sWh + w * HS + k4), 0, 0);
  }
#if __has_builtin(__builtin_amdgcn_s_wait_asynccnt)
  __builtin_amdgcn_s_wait_asynccnt(0);
#else
  asm volatile("s_wait_asynccnt 0x0" ::: "memory");
#endif
#else
  for (int idx = tid; idx < 192 * kD; idx += 128) {
    int w = idx >> 6, k = idx & 63;
    sWi[w * HS + k] = W_ih[idx];
    sWh[w * HS + k] = W_hh[idx];
  }
#endif

  const int wave = tid >> 5;
  const int lane = tid & 31;
  const int rowA = lane & 15;             // A-matrix row / B-matrix column
  const int hi   = lane >> 4;             // 0: K={k0,k0+1}, 1: K={k0+2,k0+3}
  const int bbase = (blockIdx.x * 4 + wave) * 16;
  float* H = sHal + wave * (16 * HS);

  for (int idx = lane; idx < 16 * HS; idx += 32) H[idx] = 0.0f;

  int lenr[8];
#pragma unroll
  for (int j = 0; j < 8; ++j) lenr[j] = event_lens[bbase + hi * 8 + j];

  // biases for output column (lane&15): combined for r/z, split for n
  float bRZ[8], bIN[4], bHN[4];
#pragma unroll
  for (int c = 0; c < 8; ++c) bRZ[c] = b_ih[c * 16 + rowA] + b_hh[c * 16 + rowA];
#pragma unroll
  for (int c = 0; c < 4; ++c) {
    bIN[c] = b_ih[(c + 8) * 16 + rowA];
    bHN[c] = b_hh[(c + 8) * 16 + rowA];
  }

  __syncthreads();                        // weights visible to all waves

  int eid = event_ids[bbase + rowA];      // t = 0
  for (int t = 0; t < kT; ++t) {
    const float* xrow = event_table + (size_t)eid * kD;
    float2 xa[16];
#pragma unroll
    for (int k = 0; k < 16; ++k)
      xa[k] = *reinterpret_cast<const float2*>(xrow + k * 4 + 2 * hi);
    if (t + 1 < kT) eid = event_ids[(t + 1) * kB + bbase + rowA];

    // accumulators: g[0..7] = gi+gh (r,z gates), gin/ghn split (n gate)
    v8f g[8], gin[4], ghn[4];
#pragma unroll
    for (int c = 0; c < 8; ++c) {
      v8f v;
#pragma unroll
      for (int j = 0; j < 8; ++j) v[j] = bRZ[c];
      g[c] = v;
    }
#pragma unroll
    for (int c = 0; c < 4; ++c) {
      v8f vi, vh;
#pragma unroll
      for (int j = 0; j < 8; ++j) { vi[j] = bIN[c]; vh[j] = bHN[c]; }
      gin[c] = vi; ghn[c] = vh;
    }

#pragma unroll
    for (int k0i = 0; k0i < 16; ++k0i) {
      const int k0 = k0i * 4;
      v2f a_x = { xa[k0i].x, xa[k0i].y };
      float2 ha = *reinterpret_cast<const float2*>(H + rowA * HS + k0 + 2 * hi);
      v2f a_h = { ha.x, ha.y };
#pragma unroll
      for (int c = 0; c < 8; ++c) {
        float2 wi = *reinterpret_cast<const float2*>(
            sWi + (c * 16 + rowA) * HS + k0 + 2 * hi);
        v2f b_i = { wi.x, wi.y };
        g[c] = wmma_f32x4(a_x, b_i, g[c]);
        float2 wh = *reinterpret_cast<const float2*>(
            sWh + (c * 16 + rowA) * HS + k0 + 2 * hi);
        v2f b_h = { wh.x, wh.y };
        g[c] = wmma_f32x4(a_h, b_h, g[c]);
      }
#pragma unroll
      for (int c = 0; c < 4; ++c) {
        float2 wi = *reinterpret_cast<const float2*>(
            sWi + ((c + 8) * 16 + rowA) * HS + k0 + 2 * hi);
        v2f b_i = { wi.x, wi.y };
        gin[c] = wmma_f32x4(a_x, b_i, gin[c]);
        float2 wh = *reinterpret_cast<const float2*>(
            sWh + ((c + 8) * 16 + rowA) * HS + k0 + 2 * hi);
        v2f b_h = { wh.x, wh.y };
        ghn[c] = wmma_f32x4(a_h, b_h, ghn[c]);
      }
    }

    // gates + masked hidden update (D-layout: (m = j+8*hi, col = c*16+(lane&15)))
#pragma unroll
    for (int c = 0; c < 4; ++c) {
#pragma unroll
      for (int j = 0; j < 8; ++j) {
        float rg = 1.0f / (1.0f + __expf(-g[c][j]));
        float zg = 1.0f / (1.0f + __expf(-g[c + 4][j]));
        float ng = tanhf(gin[c][j] + rg * ghn[c][j]);
        int a = (hi * 8 + j) * HS + c * 16 + rowA;
        float hold = H[a];
        float hnew = (1.0f - zg) * ng + zg * hold;
        H[a] = (t < lenr[j]) ? hnew : hold;
      }
    }
  }

  for (int idx = lane; idx < 16 * kD; idx += 32) {
    int rr = idx >> 6, cc = idx & 63;
    gru_out[(size_t)(bbase + rr) * kD + cc] = H[rr * HS + cc];
  }
}

// ---------------------------------------------------------------------------
// Distance correlation pieces
// ---------------------------------------------------------------------------
// r[b]=sum x^2, s[b]=sum x for both matrices
__global__ void rs_kernel(const float* __restrict__ X1,
                          const float* __restrict__ X2,
                          float* __restrict__ r1, float* __restrict__ s1,
                          float* __restrict__ r2, float* __restrict__ s2) {
  int b = blockIdx.x * blockDim.x + threadIdx.x;
  if (b >= kB) return;
  float rr = 0.f, ss = 0.f;
  for (int d = 0; d < kD; ++d) { float v = X1[b * kD + d]; rr += v * v; ss += v; }
  r1[b] = rr; s1[b] = ss;
  rr = 0.f; ss = 0.f;
  for (int d = 0; d < kD; ++d) { float v = X2[b * kD + d]; rr += v * v; ss += v; }
  r2[b] = rr; s2[b] = ss;
}

// One wave per 16x16 tile of D[i,j] = sqrt(max(r[i]-2*G[i,j]-2*r[j]*s[i],0)+eps)
// with G = X X^T computed by fp32 WMMA.
__global__ __launch_bounds__(256) void gram_dist_kernel(
    const float* __restrict__ X, const float* __restrict__ r,
    const float* __restrict__ s, float* __restrict__ Dout) {
  const int wave = threadIdx.x >> 5;
  const int lane = threadIdx.x & 31;
  const int rowA = lane & 15;
  const int hi   = lane >> 4;
  const int tile = blockIdx.x * 8 + wave;
  const int ibase = (tile >> 7) * 16;
  const int jbase = (tile & 127) * 16;

  v8f acc = {};
#pragma unroll 4
  for (int k0 = 0; k0 < kD; k0 += 4) {
    float2 xa = *reinterpret_cast<const float2*>(
        X + (size_t)(ibase + rowA) * kD + k0 + 2 * hi);
    v2f a = { xa.x, xa.y };
    float2 xb = *reinterpret_cast<const float2*>(
        X + (size_t)(jbase + rowA) * kD + k0 + 2 * hi);   // B[k][n] = X[jbase+n][k]
    v2f b = { xb.x, xb.y };
    acc = wmma_f32x4(a, b, acc);
  }
  const float rj = r[jbase + rowA];
#pragma unroll
  for (int j = 0; j < 8; ++j) {
    int i = ibase + hi * 8 + j;
    float val = r[i] - 2.0f * acc[j] - 2.0f * rj * s[i];
    val = fmaxf(val, 0.0f);
    Dout[(size_t)i * kB + jbase + rowA] = sqrtf(val + kEPS);
  }
}

__global__ __launch_bounds__(256) void rowsum_kernel(const float* __restrict__ Dm,
                                                     float* __restrict__ rows) {
  __shared__ float red[256];
  int i = blockIdx.x;
  float acc = 0.f;
  for (int j = threadIdx.x; j < kB; j += 256) acc += Dm[(size_t)i * kB + j];
  red[threadIdx.x] = acc;
  __syncthreads();
  for (int st = 128; st > 0; st >>= 1) {
    if (threadIdx.x < st) red[threadIdx.x] += red[threadIdx.x + st];
    __syncthreads();
  }
  if (threadIdx.x == 0) rows[i] = red[0];
}

__global__ void colsum_kernel(const float* __restrict__ Dm,
                              float* __restrict__ cols) {
  int j = blockIdx.x * blockDim.x + threadIdx.x;
  if (j >= kB) return;
  float acc = 0.f;
  for (int i = 0; i < kB; ++i) {
    if ((i & 7) == 0)   // 8KB-stride walk: help the prefetcher explicitly
      __builtin_prefetch(Dm + (size_t)(i + 16) * kB + j, 0, 0);
    acc += Dm[(size_t)i * kB + j];
  }
  cols[j] = acc;
}

__global__ __launch_bounds__(256) void total_kernel(const float* __restrict__ rows1,
                                                    const float* __restrict__ rows2,
                                                    float* __restrict__ sm) {
  __shared__ float red[256];
  float a = 0.f, b = 0.f;
  for (int i = threadIdx.x; i < kB; i += 256) { a += rows1[i]; b += rows2[i]; }
  red[threadIdx.x] = a;
  __syncthreads();
  for (int st = 128; st > 0; st >>= 1) {
    if (threadIdx.x < st) red[threadIdx.x] += red[threadIdx.x + st];
    __syncthreads();
  }
  if (threadIdx.x == 0) sm[0] = red[0];
  __syncthreads();
  red[threadIdx.x] = b;
  __syncthreads();
  for (int st = 128; st > 0; st >>= 1) {
    if (threadIdx.x < st) red[threadIdx.x] += red[threadIdx.x + st];
    __syncthreads();
  }
  if (threadIdx.x == 0) sm[1] = red[0];
}

// centered products: sm[2]+=sum c1*c2, sm[3]+=sum c1*c1, sm[4]+=sum c2*c2
__global__ __launch_bounds__(256) void cdot_kernel(
    const float* __restrict__ D1m, const float* __restrict__ D2m,
    const float* __restrict__ rows1, const float* __restrict__ cols1,
    const float* __restrict__ rows2, const float* __restrict__ cols2,
    float* __restrict__ sm) {
  __shared__ float r12[256], r11[256], r22[256];
  const float invn = 1.0f / (float)kB;
  const float mu1 = sm[0] * invn * invn;
  const float mu2 = sm[1] * invn * invn;
  float a12 = 0.f, a11 = 0.f, a22 = 0.f;
  const size_t total = (size_t)kB * kB;
  for (size_t e = (size_t)blockIdx.x * blockDim.x + threadIdx.x; e < total;
       e += (size_t)gridDim.x * blockDim.x) {
    int i = (int)(e >> 11);
    int j = (int)(e & 2047);
    float c1 = D1m[e] - cols1[j] * invn - rows1[i] * invn + mu1;
    float c2 = D2m[e] - cols2[j] * invn - rows2[i] * invn + mu2;
    a12 += c1 * c2; a11 += c1 * c1; a22 += c2 * c2;
  }
  r12[threadIdx.x] = a12; r11[threadIdx.x] = a11; r22[threadIdx.x] = a22;
  __syncthreads();
  for (int st = 128; st > 0; st >>= 1) {
    if (threadIdx.x < st) {
      r12[threadIdx.x] += r12[threadIdx.x + st];
      r11[threadIdx.x] += r11[threadIdx.x + st];
      r22[threadIdx.x] += r22[threadIdx.x + st];
    }
    __syncthreads();
  }
  if (threadIdx.x == 0) {
    atomicAdd(&sm[2], r12[0]);
    atomicAdd(&sm[3], r11[0]);
    atomicAdd(&sm[4], r22[0]);
  }
}

__global__ void finalize_kernel(const float* __restrict__ sm,
                                float* __restrict__ out_con) {
  if (threadIdx.x == 0 && blockIdx.x == 0) {
    const float n2 = (float)kB * (float)kB;
    float d12 = sqrtf(fmaxf(sm[2] / n2, 0.0f) + kEPS);
    float d11 = sqrtf(fmaxf(sm[3] / n2, 0.0f) + kEPS);
    float d22 = sqrtf(fmaxf(sm[4] / n2, 0.0f) + kEPS);
    out_con[0] = d12 / (sqrtf(d11 * d22) + 1e-10f);
  }
}

// ---------------------------------------------------------------------------
// Prediction heads: y = sigmoid((p1+p2+p3)/3), p = relu(x@Wp1+bp1)@Wp2+bp2
// ---------------------------------------------------------------------------
__global__ void predict_kernel(const int* __restrict__ ui,
                               const int* __restrict__ ii,
                               const float* __restrict__ user_table,
                               const float* __restrict__ item_table,
                               const float* __restrict__ gcn_u,
                               const float* __restrict__ gcn_i,
                               const float* __restrict__ gru_u,
                               const float* __restrict__ Wp1,
                               const float* __restrict__ bp1,
                               const float* __restrict__ Wp2,
                               const float* __restrict__ bp2,
                               float* __restrict__ y) {
  int b = blockIdx.x * blockDim.x + threadIdx.x;
  if (b >= kB) return;
  const float* up = user_table + (size_t)ui[b] * kD;
  const float* ip = item_table + (size_t)ii[b] * kD;
  const float* gu = gcn_u + (size_t)b * kD;
  const float* gi = gcn_i + (size_t)b * kD;
  const float* ru = gru_u + (size_t)b * kD;
  float h1[8], h2[8], h3[8];
#pragma unroll
  for (int j = 0; j < 8; ++j) { h1[j] = bp1[j]; h2[j] = bp1[j]; h3[j] = bp1[j]; }
  for (int d = 0; d < kD; ++d) {
    float ie = gi[d] + ip[d];
    float x1 = gu[d] * ie, x2 = ru[d] * ie, x3 = up[d] * ie;
#pragma unroll
    for (int j = 0; j < 8; ++j) {
      float w = Wp1[d * 8 + j];
      h1[j] += x1 * w; h2[j] += x2 * w; h3[j] += x3 * w;
    }
  }
  float p1 = bp2[0], p2 = bp2[0], p3 = bp2[0];
#pragma unroll
  for (int j = 0; j < 8; ++j) {
    float w = Wp2[j];
    p1 += fmaxf(h1[j], 0.f) * w;
    p2 += fmaxf(h2[j], 0.f) * w;
    p3 += fmaxf(h3[j], 0.f) * w;
  }
  float v = (p1 + p2 + p3) * (1.0f / 3.0f);
  y[b] = 1.0f / (1.0f + expf(-v));
}

// ---------------------------------------------------------------------------
// Host orchestration
// ---------------------------------------------------------------------------
extern "C" void kernel_launch(void* const* d_in, const int* in_sizes, int n_in,
                              void* d_out, int out_size, void* d_ws, size_t ws_size,
                              hipStream_t stream) {
  (void)in_sizes; (void)n_in; (void)out_size; (void)ws_size;
  const int*   user_inputs = (const int*)d_in[0];
  const int*   item_inputs = (const int*)d_in[1];
  const int*   event_ids   = (const int*)d_in[2];
  const int*   event_lens  = (const int*)d_in[3];
  const int*   graph_rows  = (const int*)d_in[4];
  const int*   graph_cols  = (const int*)d_in[5];
  const float* graph_vals  = (const float*)d_in[6];
  const float* user_table  = (const float*)d_in[7];
  const float* item_table  = (const float*)d_in[8];
  const float* event_table = (const float*)d_in[9];
  const float* W_ih        = (const float*)d_in[10];
  const float* W_hh        = (const float*)d_in[11];
  const float* b_ih        = (const float*)d_in[12];
  const float* b_hh        = (const float*)d_in[13];
  const float* Wp1         = (const float*)d_in[14];
  const float* bp1         = (const float*)d_in[15];
  const float* Wp2         = (const float*)d_in[16];
  const float* bp2         = (const float*)d_in[17];

  float* ws = (float*)d_ws;
  const size_t embN = kN * (size_t)kD;           // 9.6M floats
  float* emb_a = ws;
  float* emb_b = emb_a + embN;
  float* gcn_u = emb_b + embN;
  float* gcn_i = gcn_u + (size_t)kB * kD;
  float* gru_u = gcn_i + (size_t)kB * kD;
  float* r1 = gru_u + (size_t)kB * kD;
  float* s1 = r1 + kB;
  float* r2 = s1 + kB;
  float* s2 = r2 + kB;
  float* rows1 = s2 + kB;
  float* cols1 = rows1 + kB;
  float* rows2 = cols1 + kB;
  float* cols2 = rows2 + kB;
  float* sm = cols2 + kB;                        // 8 scalar slots
  float* D1m = emb_a;                            // alias: GCN buffers are free
  float* D2m = emb_b;                            // by the time dcor runs

  const int gB = (kB * kD + 255) / 256;          // 512 blocks for [B,64] maps

  // ---- LightGCN (mean accumulated only at gathered rows) ----
  concat_kernel<<<4096, 256, 0, stream>>>(user_table, item_table, emb_a);
  gather_init_kernel<<<gB, 256, 0, stream>>>(user_inputs, item_inputs,
                                             user_table, item_table, gcn_u, gcn_i);
  float* cur = emb_a;
  float* nxt = emb_b;
  for (int L = 0; L < 3; ++L) {
    zero_kernel<<<4096, 256, 0, stream>>>(nxt, embN);
    scatter_kernel<<<(unsigned)((kNNZ * 32 + 255) / 256), 256, 0, stream>>>(
        graph_rows, graph_cols, graph_vals, cur, nxt);
    gather_acc_kernel<<<gB, 256, 0, stream>>>(user_inputs, item_inputs, nxt,
                                              gcn_u, gcn_i,
                                              (L == 2) ? 0.25f : 1.0f);
    float* t = cur; cur = nxt; nxt = t;
  }

  // ---- zero reduction scratch (rowsums/colsums/scalars) ----
  zero_kernel<<<64, 256, 0, stream>>>(rows1, 4 * (size_t)kB + 8);

  // ---- GRU over event histories (fp32 WMMA, weights resident in LDS) ----
  constexpr size_t kGruLds = (size_t)(2 * 192 * 68 + 4 * 16 * 68) * sizeof(float);
  gru_kernel<<<kB / 64, 128, kGruLds, stream>>>(event_ids, event_lens, event_table,
                                                W_ih, W_hh, b_ih, b_hh, gru_u);

  // ---- distance correlation ----
  rs_kernel<<<(kB + 255) / 256, 256, 0, stream>>>(gcn_u, gru_u, r1, s1, r2, s2);
  gram_dist_kernel<<<(kB / 16) * (kB / 16) / 8, 256, 0, stream>>>(gcn_u, r1, s1, D1m);
  gram_dist_kernel<<<(kB / 16) * (kB / 16) / 8, 256, 0, stream>>>(gru_u, r2, s2, D2m);
  rowsum_kernel<<<kB, 256, 0, stream>>>(D1m, rows1);
  rowsum_kernel<<<kB, 256, 0, stream>>>(D2m, rows2);
  colsum_kernel<<<(kB + 255) / 256, 256, 0, stream>>>(D1m, cols1);
  colsum_kernel<<<(kB + 255) / 256, 256, 0, stream>>>(D2m, cols2);
  total_kernel<<<1, 256, 0, stream>>>(rows1, rows2, sm);
  cdot_kernel<<<1024, 256, 0, stream>>>(D1m, D2m, rows1, cols1, rows2, cols2, sm);

  // ---- prediction heads + outputs ----
  float* y_out = (float*)d_out;                 // y: [B], con_loss: d_out[B]
  predict_kernel<<<(kB + 255) / 256, 256, 0, stream>>>(
      user_inputs, item_inputs, user_table, item_table,
      gcn_u, gcn_i, gru_u, Wp1, bp1, Wp2, bp2, y_out);
  finalize_kernel<<<1, 64, 0, stream>>>(sm, y_out + kB);
}